// Sgblock_28527172780473
// MI455X (gfx1250) — compile-verified
//
#include <hip/hip_runtime.h>
#include <hip/hip_bf16.h>
#include <math.h>

typedef __attribute__((ext_vector_type(2))) float v2f;
typedef __attribute__((ext_vector_type(8))) float v8f;

#define DFEAT 128

// ---------------------------------------------------------------------------
// 1) Degree count over destination nodes (self-loop added later as +1)
// ---------------------------------------------------------------------------
__global__ void __launch_bounds__(256) sg_deg_kernel(const int* __restrict__ cols,
                                                     unsigned* __restrict__ deg,
                                                     int nE) {
  int i = blockIdx.x * blockDim.x + threadIdx.x;
  if (i < nE) atomicAdd(&deg[cols[i]], 1u);
}

// dinv[i] = rsqrt(deg_in[i] + 1)   (deg always > 0 due to self loop)
__global__ void __launch_bounds__(256) sg_dinv_kernel(const unsigned* __restrict__ deg,
                                                      float* __restrict__ dinv,
                                                      int n) {
  int i = blockIdx.x * blockDim.x + threadIdx.x;
  if (i < n) dinv[i] = rsqrtf((float)deg[i] + 1.0f);
}

// ---------------------------------------------------------------------------
// 2) Per-hop: initialize accumulator with the self-loop term
//    h_out[i,:] = dinv[i]^2 * h_in[i,:]
// ---------------------------------------------------------------------------
__global__ void __launch_bounds__(256) sg_self_kernel(const float* __restrict__ h_in,
                                                      float* __restrict__ h_out,
                                                      const float* __restrict__ dinv,
                                                      int total) {
  int i = blockIdx.x * blockDim.x + threadIdx.x;
  if (i < total) {
    float dv = dinv[i >> 7];  // i / 128
    h_out[i] = dv * dv * h_in[i];
  }
}

// ---------------------------------------------------------------------------
// 3) Edge scatter: one wave32 per edge, each lane moves a float4 (32*16B=512B)
//    h_out[col,:] += dinv[row]*dinv[col] * h_in[row,:]
// ---------------------------------------------------------------------------
__global__ void __launch_bounds__(256) sg_scatter_kernel(const float* __restrict__ h_in,
                                                         float* __restrict__ h_out,
                                                         const int* __restrict__ rows,
                                                         const int* __restrict__ cols,
                                                         const float* __restrict__ dinv,
                                                         int nE) {
  int wavesPerBlock = blockDim.x >> 5;
  int edge = blockIdx.x * wavesPerBlock + (threadIdx.x >> 5);
  if (edge >= nE) return;
  int lane = threadIdx.x & 31;

  int r = __builtin_amdgcn_readfirstlane(rows[edge]);
  int c = __builtin_amdgcn_readfirstlane(cols[edge]);
  float w = dinv[r] * dinv[c];

  const float4* src = (const float4*)(h_in + (size_t)r * DFEAT);
  float4 v = src[lane];                       // coalesced 512B row gather
  float* dst = h_out + (size_t)c * DFEAT + lane * 4;
  atomicAdd(dst + 0, w * v.x);
  atomicAdd(dst + 1, w * v.y);
  atomicAdd(dst + 2, w * v.z);
  atomicAdd(dst + 3, w * v.w);
}

// ---------------------------------------------------------------------------
// 4) Dense linear + ELU with V_WMMA_F32_16X16X4_F32.
//    out[n,j] = elu( sum_k h[n,k] * W[j,k] + b[j] )
//    One block = 16 output rows. 8 waves, wave w owns output cols [16w,16w+16).
//    A tile (16x128) staged in LDS with padded stride 132 (bank-conflict free).
// ---------------------------------------------------------------------------
#define A_STRIDE 132

__global__ void __launch_bounds__(256) sg_gemm_elu_kernel(const float* __restrict__ h,
                                                          const float* __restrict__ W,
                                                          const float* __restrict__ bias,
                                                          float* __restrict__ out) {
  __shared__ float sA[16 * A_STRIDE];

  int row0 = blockIdx.x * 16;
  int tid = threadIdx.x;

  // Cooperative load of A tile: 16 rows x 128 cols, 8 floats per thread.
  {
    int rr = tid >> 4;            // 0..15
    int ss = (tid & 15) * 8;      // 0,8,...,120
    const float* srow = h + (size_t)(row0 + rr) * DFEAT + ss;
    float4 p0 = *(const float4*)(srow);
    float4 p1 = *(const float4*)(srow + 4);
    float* drow = sA + rr * A_STRIDE + ss;
    *(float4*)(drow) = p0;
    *(float4*)(drow + 4) = p1;
  }
  __syncthreads();

  int wave = tid >> 5;            // 0..7 -> output column tile
  int lane = tid & 31;
  int j0 = wave * 16;
  int mn = lane & 15;             // A row (M) and B col (N) index
  int khalf = (lane >> 4) * 2;    // lanes 0-15: K pair {0,1}; lanes 16-31: {2,3}

  v8f c = {};
  const float* wrow = W + (size_t)(j0 + mn) * DFEAT;  // B column N = row of W

#pragma unroll
  for (int k = 0; k < DFEAT; k += 4) {
    v2f a, b;
    a.x = sA[mn * A_STRIDE + k + khalf];
    a.y = sA[mn * A_STRIDE + k + khalf + 1];
    float2 bw = *(const float2*)(wrow + k + khalf);
    b.x = bw.x;
    b.y = bw.y;
    // D = A(16x4) * B(4x16) + C   -> v_wmma_f32_16x16x4_f32
    c = __builtin_amdgcn_wmma_f32_16x16x4_f32(false, a, false, b,
                                              (short)0, c, false, false);
  }

  float bj = bias[j0 + mn];
  int mbase = (lane >> 4) * 8;    // VGPR v: lanes 0-15 -> M=v, lanes 16-31 -> M=v+8
#pragma unroll
  for (int v = 0; v < 8; v++) {
    float x = c[v] + bj;
    float y = x > 0.0f ? x : (__expf(x) - 1.0f);  // ELU(alpha=1)
    out[(size_t)(row0 + mbase + v) * DFEAT + (j0 + mn)] = y;
  }
}

// ---------------------------------------------------------------------------
// Host-side launch
// ---------------------------------------------------------------------------
extern "C" void kernel_launch(void* const* d_in, const int* in_sizes, int n_in,
                              void* d_out, int out_size, void* d_ws, size_t ws_size,
                              hipStream_t stream) {
  const float* x  = (const float*)d_in[0];
  const int*   ei = (const int*)d_in[1];
  const float* W  = (const float*)d_in[2];
  const float* b  = (const float*)d_in[3];

  const int N = in_sizes[0] / DFEAT;     // 100000
  const int E = in_sizes[1] / 2;         // 3200000
  const int* rows = ei;                  // edge_index[0] = source
  const int* cols = ei + E;              // edge_index[1] = destination

  float* out = (float*)d_out;

  // Workspace layout: [h_tmp: N*128 f32][dinv: N f32][deg: N u32]
  float*    h_tmp = (float*)d_ws;
  float*    dinv  = h_tmp + (size_t)N * DFEAT;
  unsigned* deg   = (unsigned*)(dinv + N);

  // --- degree + normalization ---
  hipMemsetAsync(deg, 0, (size_t)N * sizeof(unsigned), stream);
  sg_deg_kernel<<<(E + 255) / 256, 256, 0, stream>>>(cols, deg, E);
  sg_dinv_kernel<<<(N + 255) / 256, 256, 0, stream>>>(deg, dinv, N);

  const int total = N * DFEAT;
  const int selfBlocks = (total + 255) / 256;
  const int scatBlocks = (E + 7) / 8;    // 8 waves/block, one edge per wave

  // --- hop 1: x -> h_tmp ---
  sg_self_kernel<<<selfBlocks, 256, 0, stream>>>(x, h_tmp, dinv, total);
  sg_scatter_kernel<<<scatBlocks, 256, 0, stream>>>(x, h_tmp, rows, cols, dinv, E);

  // --- hop 2: h_tmp -> out (used as scratch) ---
  sg_self_kernel<<<selfBlocks, 256, 0, stream>>>(h_tmp, out, dinv, total);
  sg_scatter_kernel<<<scatBlocks, 256, 0, stream>>>(h_tmp, out, rows, cols, dinv, E);

  // --- hop 3: out -> h_tmp ---
  sg_self_kernel<<<selfBlocks, 256, 0, stream>>>(out, h_tmp, dinv, total);
  sg_scatter_kernel<<<scatBlocks, 256, 0, stream>>>(out, h_tmp, rows, cols, dinv, E);

  // --- linear + ELU via f32 WMMA: h_tmp @ W^T + b -> out ---
  sg_gemm_elu_kernel<<<N / 16, 256, 0, stream>>>(h_tmp, W, b, out);
}